// LightweightGNN_70111046140372
// MI455X (gfx1250) — compile-verified
//
#include <hip/hip_runtime.h>
#include <hip/hip_bf16.h>

// ---------------------------------------------------------------------------
// 2-layer GAT on MI455X (gfx1250, wave32).
//  - Layer-1 linear uses V_WMMA_F32_16X16X4_F32 (fp32-exact tensor op).
//  - Edge softmax/aggregation: 3 scatter passes; 51MB accumulator is
//    L2-resident (192MB L2) so fp32 atomics resolve at L2.
// ---------------------------------------------------------------------------

#define NN   100000
#define EE   1600000
#define IN_C 64
#define HID  32
#define HEADS 4
#define HW1  (HEADS*HID)      // 128
#define NEG_SLOPE 0.2f
#define BN_EPS 1e-5f

typedef __attribute__((ext_vector_type(2))) float v2f;
typedef __attribute__((ext_vector_type(8))) float v8f;

// ---------------- init: zero accumulators, -inf maxima, zero d_out ----------
__global__ void k_init(float* __restrict__ hagg, float* __restrict__ den1,
                       float* __restrict__ emax1, float* __restrict__ den2,
                       float* __restrict__ emax2, float* __restrict__ out) {
  int i = blockIdx.x * blockDim.x + threadIdx.x;
  const float ninf = __int_as_float(0xFF800000);
  if (i < NN * HW1) hagg[i] = 0.0f;
  if (i < NN * HEADS) { den1[i] = 0.0f; emax1[i] = ninf; }
  if (i < NN) { den2[i] = 0.0f; emax2[i] = ninf; out[i] = 0.0f; }
}

// ---------------- layer-1 GEMM: h1 = x @ W1 via WMMA f32 16x16x4 ------------
// block = 256 threads = 8 waves; block handles a 16-row slab of x (via LDS),
// wave w computes the 16x16 output tile at columns [16w, 16w+16).
__global__ void k_gemm1(const float* __restrict__ x, const float* __restrict__ W1,
                        float* __restrict__ h1) {
  __shared__ float xs[16 * IN_C];
  const int tid  = threadIdx.x;
  const int slab = blockIdx.x;                // 16-row slab, N/16 = 6250 exact
  // cooperative coalesced load of the 16x64 slab (1024 floats, 4 per thread)
  const float4 xin = *(const float4*)(x + (size_t)slab * 16 * IN_C + tid * 4);
  *(float4*)(xs + tid * 4) = xin;
  __syncthreads();

  const int lane  = tid & 31;
  const int wave  = tid >> 5;
  const int col0  = wave * 16;
  const int m     = lane & 15;                // M (A) / N (B,store)
  const int khalf = lane >> 4;                // 0: K{0,1}, 1: K{2,3}

  v8f c = {};
  #pragma unroll
  for (int k0 = 0; k0 < IN_C; k0 += 4) {
    const int ka = k0 + 2 * khalf;
    v2f a, b;
    a.x = xs[m * IN_C + ka];                  // A(m, ka)
    a.y = xs[m * IN_C + ka + 1];              // A(m, ka+1)
    b.x = W1[ka * HW1 + col0 + m];            // B(ka, n)
    b.y = W1[(ka + 1) * HW1 + col0 + m];      // B(ka+1, n)
    c = __builtin_amdgcn_wmma_f32_16x16x4_f32(
            false, a, false, b, (short)0, c, false, false);
  }
  // C/D layout: VGPR i -> M = i + 8*khalf, N = lane&15
  const size_t row0 = (size_t)slab * 16 + 8 * khalf;
  #pragma unroll
  for (int i = 0; i < 8; ++i)
    h1[(row0 + i) * HW1 + col0 + m] = c[i];
}

// ---------------- per-node attention terms: [N, HEADS] ----------------------
__global__ void k_alpha1(const float* __restrict__ h1,
                         const float* __restrict__ att_s,
                         const float* __restrict__ att_d,
                         float* __restrict__ as1, float* __restrict__ ad1) {
  int idx = blockIdx.x * blockDim.x + threadIdx.x;   // n*HEADS + h
  if (idx >= NN * HEADS) return;
  const int n = idx >> 2, h = idx & 3;
  const float* hp = h1 + (size_t)n * HW1 + h * HID;
  const float* sp = att_s + h * HID;
  const float* dp = att_d + h * HID;
  float s = 0.f, d = 0.f;
  #pragma unroll
  for (int c = 0; c < HID; ++c) { float v = hp[c]; s += v * sp[c]; d += v * dp[c]; }
  as1[idx] = s; ad1[idx] = d;
}

// ---------------- float atomic max via sign-split int atomics ---------------
__device__ __forceinline__ void atomicMaxF32(float* addr, float v) {
  if (v >= 0.0f) atomicMax((int*)addr, __float_as_int(v));
  else           atomicMin((unsigned int*)addr, __float_as_uint(v));
}

__device__ __forceinline__ float leaky(float v) {
  return v > 0.0f ? v : NEG_SLOPE * v;
}

__device__ __forceinline__ void edge_sd(const int* __restrict__ ei, int e,
                                        int& s, int& d) {
  if (e < EE) { s = ei[e]; d = ei[EE + e]; } else { s = d = e - EE; }
}

// ---------------- edge pass A: segment max of logits ------------------------
__global__ void k_edge_max1(const int* __restrict__ ei,
                            const float* __restrict__ as1,
                            const float* __restrict__ ad1,
                            float* __restrict__ emax1) {
  int idx = blockIdx.x * blockDim.x + threadIdx.x;   // edge*HEADS + h
  if (idx >= (EE + NN) * HEADS) return;
  const int e = idx >> 2, h = idx & 3;
  int s, d; edge_sd(ei, e, s, d);
  atomicMaxF32(&emax1[d * HEADS + h], leaky(as1[s * HEADS + h] + ad1[d * HEADS + h]));
}

// ---------------- edge pass B: segment sum of exp ---------------------------
__global__ void k_edge_sum1(const int* __restrict__ ei,
                            const float* __restrict__ as1,
                            const float* __restrict__ ad1,
                            const float* __restrict__ emax1,
                            float* __restrict__ den1) {
  int idx = blockIdx.x * blockDim.x + threadIdx.x;
  if (idx >= (EE + NN) * HEADS) return;
  const int e = idx >> 2, h = idx & 3;
  int s, d; edge_sd(ei, e, s, d);
  const float v = leaky(as1[s * HEADS + h] + ad1[d * HEADS + h]);
  atomicAdd(&den1[d * HEADS + h], __expf(v - emax1[d * HEADS + h]));
}

// ---------------- edge pass C: weighted aggregation (wave per edge) ---------
__global__ void k_edge_agg1(const int* __restrict__ ei,
                            const float* __restrict__ as1,
                            const float* __restrict__ ad1,
                            const float* __restrict__ emax1,
                            const float* __restrict__ den1,
                            const float* __restrict__ h1,
                            float* __restrict__ hagg) {
  const int lane = threadIdx.x & 31;
  const int e = blockIdx.x * (blockDim.x >> 5) + (threadIdx.x >> 5);
  if (e >= EE + NN) return;
  int s, d; edge_sd(ei, e, s, d);
  const int h = lane >> 3;                           // 4 channels/lane
  const float v = leaky(as1[s * HEADS + h] + ad1[d * HEADS + h]);
  const float alpha = __expf(v - emax1[d * HEADS + h]) /
                      (den1[d * HEADS + h] + 1e-16f);
  const float4 hv = *(const float4*)(h1 + (size_t)s * HW1 + lane * 4);
  float* outp = hagg + (size_t)d * HW1 + lane * 4;
  atomicAdd(outp + 0, hv.x * alpha);
  atomicAdd(outp + 1, hv.y * alpha);
  atomicAdd(outp + 2, hv.z * alpha);
  atomicAdd(outp + 3, hv.w * alpha);
}

// ---------------- bias + BatchNorm(eval) + ELU (in place) -------------------
__global__ void k_bn_elu(float* __restrict__ hagg,
                         const float* __restrict__ b1,
                         const float* __restrict__ g, const float* __restrict__ be,
                         const float* __restrict__ mu, const float* __restrict__ var) {
  int idx = blockIdx.x * blockDim.x + threadIdx.x;
  if (idx >= NN * HW1) return;
  const int c = idx & (HW1 - 1);
  float v = hagg[idx] + b1[c];
  v = (v - mu[c]) * __frsqrt_rn(var[c] + BN_EPS) * g[c] + be[c];
  hagg[idx] = v > 0.0f ? v : __expf(v) - 1.0f;
}

// ---------------- layer-2 linear: t[n] = h2[n,:] . W2 (wave per node) -------
__global__ void k_lin2(const float* __restrict__ h2,
                       const float* __restrict__ W2, float* __restrict__ t) {
  const int lane = threadIdx.x & 31;
  const int n = blockIdx.x * (blockDim.x >> 5) + (threadIdx.x >> 5);
  if (n >= NN) return;
  const float4 hv = *(const float4*)(h2 + (size_t)n * HW1 + lane * 4);
  const float4 wv = *(const float4*)(W2 + lane * 4);
  float p = hv.x * wv.x + hv.y * wv.y + hv.z * wv.z + hv.w * wv.w;
  #pragma unroll
  for (int off = 16; off > 0; off >>= 1) p += __shfl_xor(p, off, 32);
  if (lane == 0) t[n] = p;
}

// ---------------- layer-2 edge passes (single head, scalar) -----------------
__global__ void k_edge_max2(const int* __restrict__ ei, const float* __restrict__ t,
                            const float* __restrict__ asw, const float* __restrict__ adw,
                            float* __restrict__ emax2) {
  int e = blockIdx.x * blockDim.x + threadIdx.x;
  if (e >= EE + NN) return;
  int s, d; edge_sd(ei, e, s, d);
  atomicMaxF32(&emax2[d], leaky(t[s] * asw[0] + t[d] * adw[0]));
}

__global__ void k_edge_sum2(const int* __restrict__ ei, const float* __restrict__ t,
                            const float* __restrict__ asw, const float* __restrict__ adw,
                            const float* __restrict__ emax2, float* __restrict__ den2) {
  int e = blockIdx.x * blockDim.x + threadIdx.x;
  if (e >= EE + NN) return;
  int s, d; edge_sd(ei, e, s, d);
  const float v = leaky(t[s] * asw[0] + t[d] * adw[0]);
  atomicAdd(&den2[d], __expf(v - emax2[d]));
}

__global__ void k_edge_agg2(const int* __restrict__ ei, const float* __restrict__ t,
                            const float* __restrict__ asw, const float* __restrict__ adw,
                            const float* __restrict__ emax2, const float* __restrict__ den2,
                            float* __restrict__ out) {
  int e = blockIdx.x * blockDim.x + threadIdx.x;
  if (e >= EE + NN) return;
  int s, d; edge_sd(ei, e, s, d);
  const float v = leaky(t[s] * asw[0] + t[d] * adw[0]);
  const float alpha = __expf(v - emax2[d]) / (den2[d] + 1e-16f);
  atomicAdd(&out[d], t[s] * alpha);
}

__global__ void k_bias2(float* __restrict__ out, const float* __restrict__ b2) {
  int n = blockIdx.x * blockDim.x + threadIdx.x;
  if (n < NN) out[n] += b2[0];
}

// ---------------------------------------------------------------------------
extern "C" void kernel_launch(void* const* d_in, const int* in_sizes, int n_in,
                              void* d_out, int out_size, void* d_ws, size_t ws_size,
                              hipStream_t stream) {
  const float* x     = (const float*)d_in[0];
  const int*   ei    = (const int*)  d_in[1];
  const float* W1    = (const float*)d_in[2];
  const float* attS1 = (const float*)d_in[3];
  const float* attD1 = (const float*)d_in[4];
  const float* b1    = (const float*)d_in[5];
  const float* W2    = (const float*)d_in[6];
  const float* attS2 = (const float*)d_in[7];
  const float* attD2 = (const float*)d_in[8];
  const float* b2    = (const float*)d_in[9];
  const float* bn_g  = (const float*)d_in[10];
  const float* bn_b  = (const float*)d_in[11];
  const float* bn_m  = (const float*)d_in[12];
  const float* bn_v  = (const float*)d_in[13];
  float* out = (float*)d_out;

  // workspace layout: N*(128+128+4+4+4+4+1+1+1) floats ~= 110 MB
  float* ws    = (float*)d_ws;
  float* h1    = ws;                       // N*128
  float* hagg  = h1    + (size_t)NN * HW1; // N*128 (becomes h2 in place)
  float* as1   = hagg  + (size_t)NN * HW1; // N*4
  float* ad1   = as1   + (size_t)NN * HEADS;
  float* emax1 = ad1   + (size_t)NN * HEADS;
  float* den1  = emax1 + (size_t)NN * HEADS;
  float* t2    = den1  + (size_t)NN * HEADS; // N
  float* emax2 = t2    + (size_t)NN;
  float* den2  = emax2 + (size_t)NN;

  const int ET = EE + NN;                  // edges incl. self-loops

  k_init     <<<(NN * HW1 + 255) / 256, 256, 0, stream>>>(hagg, den1, emax1, den2, emax2, out);
  k_gemm1    <<<NN / 16, 256, 0, stream>>>(x, W1, h1);
  k_alpha1   <<<(NN * HEADS + 255) / 256, 256, 0, stream>>>(h1, attS1, attD1, as1, ad1);
  k_edge_max1<<<(ET * HEADS + 255) / 256, 256, 0, stream>>>(ei, as1, ad1, emax1);
  k_edge_sum1<<<(ET * HEADS + 255) / 256, 256, 0, stream>>>(ei, as1, ad1, emax1, den1);
  k_edge_agg1<<<(ET + 7) / 8, 256, 0, stream>>>(ei, as1, ad1, emax1, den1, h1, hagg);
  k_bn_elu   <<<(NN * HW1 + 255) / 256, 256, 0, stream>>>(hagg, b1, bn_g, bn_b, bn_m, bn_v);
  k_lin2     <<<(NN + 7) / 8, 256, 0, stream>>>(hagg, W2, t2);
  k_edge_max2<<<(ET + 255) / 256, 256, 0, stream>>>(ei, t2, attS2, attD2, emax2);
  k_edge_sum2<<<(ET + 255) / 256, 256, 0, stream>>>(ei, t2, attS2, attD2, emax2, den2);
  k_edge_agg2<<<(ET + 255) / 256, 256, 0, stream>>>(ei, t2, attS2, attD2, emax2, den2, out);
  k_bias2    <<<(NN + 255) / 256, 256, 0, stream>>>(out, b2);
}